// SynthesisLayer_33629593928165
// MI455X (gfx1250) — compile-verified
//
#include <hip/hip_runtime.h>

typedef __attribute__((ext_vector_type(2))) float v2f;
typedef __attribute__((ext_vector_type(8))) float v8f;

#define NB    8
#define CIN   128
#define COUT  128
#define SIN   128
#define SS    (SIN*SIN)      // 16384
#define SCV   130            // conv output spatial: 128 + 2*2 - 3 + 1
#define NPIX  (SCV*SCV)      // 16900
#define KTOT  (CIN*9)        // 1152
#define SOUT  128
#define TILE  32
#define PT    (2*TILE+11)    // 75: post-upsample patch extent per tile
#define NS    44             // conv-output patch extent per tile

// ---------------------------------------------------------------------------
// Kernel 1: 3x3 conv as implicit GEMM with V_WMMA_F32_16X16X4_F32.
// K ordering: k = t*128 + ci  (t = ky*3+kx).  For fixed t, the im2col
// coordinates and validity are invariant over ci, so the ci loop is pure
// loads (immediate offsets) + mask-muls + WMMA: no divides, no EXEC branches.
// Wave = 16 Cout x 64 pixels (4 N-subtiles share each A fragment);
// block = 8 waves = all 128 Cout for one 64-pixel group.
// A frag (16x4): lane L -> row mbase+(L&15); lanes 0-15 K=k0,k0+1,
//   lanes 16-31 K=k0+2,k0+3 (per ISA table).  B frag mirrored.
// ---------------------------------------------------------------------------
__global__ __launch_bounds__(256)
void conv3x3_bias_wmma(const float* __restrict__ X, const float* __restrict__ W,
                       const float* __restrict__ bconv, const float* __restrict__ bias,
                       float* __restrict__ Y)
{
    const int lane  = threadIdx.x & 31;
    const int wave  = threadIdx.x >> 5;
    const int hi    = lane >> 4;          // 0: lanes 0-15, 1: lanes 16-31
    const int ln    = lane & 15;
    const int b     = blockIdx.y;
    const int nbase = blockIdx.x << 6;    // 64 pixels per block
    const int mbase = wave << 4;          // 8 waves cover Cout = 128

    int  n[4], oyv[4], oxv[4];
    bool nv[4];
#pragma unroll
    for (int p = 0; p < 4; ++p) {
        n[p]  = nbase + (p << 4) + ln;
        nv[p] = n[p] < NPIX;
        const int q = nv[p] ? n[p] : 0;
        oyv[p] = q / SCV;
        oxv[p] = q - oyv[p] * SCV;
    }

    // Fold the lane-half K offset (+2 in k => ci+2) into the base pointers:
    // W addr = (mbase+ln)*1152 + ci*9 + t ; X addr = ci*SS + iy*SIN + ix.
    const float* Xb = X + (size_t)b * CIN * SS + hi * 2 * SS;
    const float* Wr = W + (size_t)(mbase + ln) * KTOT + hi * 2 * 9;

    v8f acc[4] = {{}, {}, {}, {}};

#pragma unroll 1
    for (int t = 0; t < 9; ++t) {
        const int ky = t / 3;
        const int kx = t - 3 * ky;
        const float* aB = Wr + t;

        const float* xp[4];
        float msk[4];
#pragma unroll
        for (int p = 0; p < 4; ++p) {
            const int iy = oyv[p] + ky - 2;
            const int ix = oxv[p] + kx - 2;
            const bool v = nv[p] && (unsigned)iy < (unsigned)SIN
                                 && (unsigned)ix < (unsigned)SIN;
            msk[p] = v ? 1.0f : 0.0f;                 // one cndmask per tap
            xp[p]  = Xb + (v ? iy * SIN + ix : 0);    // clamped: always safe
        }

#pragma unroll 4
        for (int cb = 0; cb < CIN; cb += 4) {
            v2f a;                                    // A: strides of 9 floats
            a.x = aB[cb * 9];
            a.y = aB[cb * 9 + 9];

            v2f b0, b1, b2, b3;                       // B: strides of SS floats
            b0.x = msk[0] * xp[0][cb * SS];
            b0.y = msk[0] * xp[0][cb * SS + SS];
            b1.x = msk[1] * xp[1][cb * SS];
            b1.y = msk[1] * xp[1][cb * SS + SS];
            b2.x = msk[2] * xp[2][cb * SS];
            b2.y = msk[2] * xp[2][cb * SS + SS];
            b3.x = msk[3] * xp[3][cb * SS];
            b3.y = msk[3] * xp[3][cb * SS + SS];

            acc[0] = __builtin_amdgcn_wmma_f32_16x16x4_f32(false, a, false, b0,
                                                           (short)0, acc[0], false, false);
            acc[1] = __builtin_amdgcn_wmma_f32_16x16x4_f32(false, a, false, b1,
                                                           (short)0, acc[1], false, false);
            acc[2] = __builtin_amdgcn_wmma_f32_16x16x4_f32(false, a, false, b2,
                                                           (short)0, acc[2], false, false);
            acc[3] = __builtin_amdgcn_wmma_f32_16x16x4_f32(false, a, false, b3,
                                                           (short)0, acc[3], false, false);
        }
    }

    // Fused bias adds; C/D layout: VGPR v -> M = v + 8*hi, N = lane&15.
    float bb[8];
#pragma unroll
    for (int v = 0; v < 8; ++v) {
        const int co = mbase + (hi << 3) + v;
        bb[v] = bconv[co] + bias[co];
    }
#pragma unroll
    for (int p = 0; p < 4; ++p) {
        if (!nv[p]) continue;
        float* dst = Y + ((size_t)b * COUT + (mbase + (hi << 3))) * NPIX
                       + oyv[p] * SCV + oxv[p];
#pragma unroll
        for (int v = 0; v < 8; ++v)
            dst[v * NPIX] = acc[p][v] + bb[v];
    }
}

// ---------------------------------------------------------------------------
// Kernel 2: fully fused filtered leaky-ReLU, one 32x32 output tile per block.
//   conv-out patch T0[44x44] -> upH U[75x44] -> upW+lrelu P[75x75]
//   -> downH DH[32x75] -> downW -> out[32x32]
// Up pass (up=2, taps=12, pad 9/9): out[r] = sum_j frev[j]*xd[r+j-9], xd zero-
// stuffed; parity of j selects the 6 live taps.  Down: out[r]=sum_j frev[j]*x[2r+j].
// frev[j]=f[11-j]; up filter carries gain 2/pass; lrelu slope .2, x sqrt(2).
// ---------------------------------------------------------------------------
__global__ __launch_bounds__(256)
void filtered_lrelu_tile(const float* __restrict__ T, const float* __restrict__ fup,
                         const float* __restrict__ fdn, float* __restrict__ O)
{
    __shared__ float ldsA[NS * NS];      // T0
    __shared__ float ldsP[PT * PT];      // P
    __shared__ float ldsB[PT * NS];      // U, then DH (2400 <= 3300)
    __shared__ float FU[12], FD[12];

    const int tid = threadIdx.x;
    const int bc  = blockIdx.y;                 // batch*channel, 0..1023
    const int r0  = (blockIdx.x >> 2) * TILE;
    const int c0  = (blockIdx.x & 3)  * TILE;
    const int gy0 = 2 * r0;
    const int gx0 = 2 * c0;
    const int s0  = r0 - 5;
    const int sx0 = c0 - 5;

    if (tid < 12)      FU[tid]      = fup[11 - tid] * 2.0f;
    else if (tid < 24) FD[tid - 12] = fdn[23 - tid];

    const float* src = T + (size_t)bc * NPIX;
    for (int idx = tid; idx < NS * NS; idx += 256) {
        const int ls = idx / NS, lx = idx - ls * NS;
        const int s  = s0 + ls, gx = sx0 + lx;
        float v = 0.0f;
        if ((unsigned)s < (unsigned)SCV && (unsigned)gx < (unsigned)SCV)
            v = src[s * SCV + gx];
        ldsA[idx] = v;
    }
    __syncthreads();

    for (int idx = tid; idx < PT * NS; idx += 256) {   // upH
        const int ii = idx / NS, xx = idx - ii * NS;
        const int i  = gy0 + ii;
        int j = (i + 1) & 1;
        float a = 0.0f;
#pragma unroll
        for (int u = 0; u < 6; ++u, j += 2) {
            const int ls = ((i + j - 9) >> 1) - s0;    // in [0,NS)
            a += FU[j] * ldsA[ls * NS + xx];
        }
        ldsB[idx] = a;
    }
    __syncthreads();

    for (int idx = tid; idx < PT * PT; idx += 256) {   // upW + lrelu
        const int ii = idx / PT, xo = idx - ii * PT;
        const int jx = gx0 + xo;
        int j = (jx + 1) & 1;
        float a = 0.0f;
#pragma unroll
        for (int u = 0; u < 6; ++u, j += 2) {
            const int lsx = ((jx + j - 9) >> 1) - sx0; // in [0,NS)
            a += FU[j] * ldsB[ii * NS + lsx];
        }
        a = (a >= 0.0f ? a : 0.2f * a) * 1.41421356237f;
        ldsP[idx] = a;
    }
    __syncthreads();

    for (int idx = tid; idx < TILE * PT; idx += 256) { // downH
        const int rr = idx / PT, xx = idx - rr * PT;
        float a = 0.0f;
#pragma unroll
        for (int j = 0; j < 12; ++j)
            a += FD[j] * ldsP[(2 * rr + j) * PT + xx];
        ldsB[idx] = a;
    }
    __syncthreads();

    float* dst = O + (size_t)bc * (SOUT * SOUT);
    for (int idx = tid; idx < TILE * TILE; idx += 256) { // downW
        const int rr = idx / TILE, cc = idx - rr * TILE;
        float a = 0.0f;
#pragma unroll
        for (int j = 0; j < 12; ++j)
            a += FD[j] * ldsB[rr * PT + 2 * cc + j];
        dst[(r0 + rr) * SOUT + c0 + cc] = a;
    }
}

// ---------------------------------------------------------------------------
// Launch: conv -> tmp0 (69 MB in d_ws), then fused filtered-lrelu -> d_out.
// ---------------------------------------------------------------------------
extern "C" void kernel_launch(void* const* d_in, const int* in_sizes, int n_in,
                              void* d_out, int out_size, void* d_ws, size_t ws_size,
                              hipStream_t stream)
{
    const float* x   = (const float*)d_in[0];   // [8,128,128,128]
    const float* w   = (const float*)d_in[1];   // [128,128,3,3]
    const float* bcv = (const float*)d_in[2];   // [128]
    const float* bs  = (const float*)d_in[3];   // [128]
    const float* fup = (const float*)d_in[4];   // [12]
    const float* fdn = (const float*)d_in[5];   // [12]
    float*       out = (float*)d_out;           // [8,128,128,128]
    float*       tmp0 = (float*)d_ws;           // [8,128,130,130]

    dim3 gConv((NPIX + 63) / 64, NB);           // 265 pixel-groups x 8 batches
    conv3x3_bias_wmma<<<gConv, 256, 0, stream>>>(x, w, bcv, bs, tmp0);

    dim3 gF((SOUT / TILE) * (SOUT / TILE), NB * COUT);  // 16 tiles x 1024 images
    filtered_lrelu_tile<<<gF, 256, 0, stream>>>(tmp0, fup, fdn, out);
}